// BasicBlock_45191645888983
// MI455X (gfx1250) — compile-verified
//
#include <hip/hip_runtime.h>
#include <hip/hip_bf16.h>
#include <stdint.h>

typedef __attribute__((ext_vector_type(2))) float v2f;
typedef __attribute__((ext_vector_type(4))) float v4f;
typedef __attribute__((ext_vector_type(8))) float v8f;

#define BN_EPS 1e-5f

// ---- GEMM tile config ----
#define TR 128        // rows per block (8 waves x 16)
#define TC 64         // cols per block (4 WMMA tiles per wave)
#define KC 64         // K chunk staged in LDS
#define AP (KC + 2)   // padded LDS pitch (floats) -> conflict-free b64 reads

enum { MODE_PLAIN = 0, MODE_NEGINF = 1, MODE_GATHER = 2 };

struct GatherArgs {
  const float* lastF0; const float* lastC0; const float* curC0; const int* edge0;
  const float* lastF1; const float* lastC1; const float* curC1; const int* edge1;
  int E; int F;
};

// Fused GEMM: out = relu(A @ W + bias); also accumulates per-column sum/sumsq
// for training-mode BatchNorm statistics.
// A-fragment layout (V_WMMA_F32_16X16X4_F32): lanes 0-15 hold A[m][k,k+1],
// lanes 16-31 hold A[m][k+2,k+3] (m = lane&15). B symmetric over K.
template <int MODE>
__global__ __launch_bounds__(256)
void gemm_relu_stats(const float* __restrict__ A, const float* __restrict__ W,
                     const float* __restrict__ bias, float* __restrict__ out,
                     float* __restrict__ gsum, float* __restrict__ gsq,
                     int rows, int K, int N, GatherArgs ga) {
  __shared__ float As[TR][AP];   // A tile, row-major
  __shared__ float Bs[TC][AP];   // B tile, transposed: Bs[n][k]
  __shared__ float sredS[TC];
  __shared__ float sredQ[TC];

  const int tid  = threadIdx.x;
  const int wave = tid >> 5;
  const int lane = tid & 31;
  const int ln   = lane & 15;
  const int hi   = lane >> 4;      // half-wave: 0 or 1
  const int koff = hi << 1;        // K offset of this half (0 or 2)
  const int row0 = blockIdx.x * TR;
  const int col0 = blockIdx.y * TC;

  if (tid < TC) { sredS[tid] = 0.f; sredQ[tid] = 0.f; }

  v8f acc[4];
#pragma unroll
  for (int t = 0; t < 4; ++t)
#pragma unroll
    for (int v = 0; v < 8; ++v) acc[t][v] = 0.f;

  for (int kc = 0; kc < K; kc += KC) {
    // ---- stage A tile (128 x 64) ----
    for (int i = 0; i < (TR * KC) / 256; ++i) {
      int idx = tid + i * 256;
      int rl = idx >> 6, kl = idx & 63;
      int r = row0 + rl, k = kc + kl;
      float val = 0.f;
      if (MODE == MODE_GATHER) {
        if (r < rows) {
          int half = (r >= ga.E) ? 1 : 0;
          int re = r - (half ? ga.E : 0);
          const int* eg = half ? ga.edge1 : ga.edge0;
          int ci = eg[re];
          int li = eg[ga.E + re];
          if (k < ga.F) {
            val = (half ? ga.lastF1 : ga.lastF0)[(long)li * ga.F + k];
          } else {
            int d = k - ga.F;
            val = (half ? ga.lastC1 : ga.lastC0)[li * 3 + d]
                - (half ? ga.curC1  : ga.curC0 )[ci * 3 + d];
          }
        }
      } else {
        val = (r < rows) ? A[(long)r * K + k] : 0.f;
        if (MODE == MODE_NEGINF && __float_as_uint(val) == 0xFF800000u)
          val = 0.f;  // empty segment: -inf -> 0 (matches reference)
      }
      As[rl][kl] = val;
    }
    // ---- stage B tile transposed (64 cols x 64 k) ----
    for (int i = 0; i < (TC * KC) / 256; ++i) {
      int idx = tid + i * 256;
      int kl = idx >> 6, nl = idx & 63;
      Bs[nl][kl] = W[(long)(kc + kl) * N + col0 + nl];
    }
    __syncthreads();

    const int rl = wave * 16 + ln;
#pragma unroll
    for (int k = 0; k < KC; k += 4) {
      v2f a = *(const v2f*)&As[rl][k + koff];
#pragma unroll
      for (int t = 0; t < 4; ++t) {
        v2f b = *(const v2f*)&Bs[t * 16 + ln][k + koff];
        acc[t] = __builtin_amdgcn_wmma_f32_16x16x4_f32(
            false, a, false, b, (short)0, acc[t], false, false);
      }
    }
    __syncthreads();
  }

  // ---- epilogue: bias + relu + store + per-column partial stats ----
#pragma unroll
  for (int t = 0; t < 4; ++t) {
    const int n = col0 + t * 16 + ln;
    const float bn = bias[n];
    float s = 0.f, q = 0.f;
#pragma unroll
    for (int v = 0; v < 8; ++v) {
      int r = row0 + wave * 16 + hi * 8 + v;  // C layout: lanes>=16 hold M=8..15
      float x = fmaxf(acc[t][v] + bn, 0.f);
      if (r < rows) {
        out[(long)r * N + n] = x;
        s += x; q += x * x;
      }
    }
    s += __shfl_xor(s, 16, 32);
    q += __shfl_xor(q, 16, 32);
    if (hi == 0) {
      atomicAdd(&sredS[t * 16 + ln], s);   // ds_add_f32
      atomicAdd(&sredQ[t * 16 + ln], q);
    }
  }
  __syncthreads();
  if (tid < TC) {
    atomicAdd(&gsum[col0 + tid], sredS[tid]);
    atomicAdd(&gsq[col0 + tid], sredQ[tid]);
  }
}

// agg := -inf (bit pattern), stats := 0
__global__ void init_kernel(unsigned* agg, long aggN, float* stats, long statsN) {
  long i = (long)blockIdx.x * blockDim.x + threadIdx.x;
  if (i < aggN) agg[i] = 0xFF800000u;
  else if (i < aggN + statsN) stats[i - aggN] = 0.f;
}

// Fold BN affine of previous layer into next layer's weights:
// a[k] = g[k]*rsqrt(var+eps), c[k] = be[k]-m[k]*a[k]
// Wf[k][n] = a[k]*W[k][n];  bf[n] = b[n] + sum_k c[k]*W[k][n]
__global__ __launch_bounds__(256)
void finalize_fold(const float* __restrict__ sum, const float* __restrict__ sq,
                   float invRows, const float* __restrict__ g,
                   const float* __restrict__ be, const float* __restrict__ W,
                   const float* __restrict__ b, float* __restrict__ Wf,
                   float* __restrict__ bf, int K, int N) {
  __shared__ float sa[256], sc[256];
  int tid = threadIdx.x;
  if (tid < K) {
    float m = sum[tid] * invRows;
    float v = fmaxf(sq[tid] * invRows - m * m, 0.f);  // biased var
    float s = rsqrtf(v + BN_EPS) * g[tid];
    sa[tid] = s;
    sc[tid] = be[tid] - m * s;
  }
  __syncthreads();
  for (int n = tid; n < N; n += 256) {
    float acc = b[n];
    for (int k = 0; k < K; ++k) {
      float w = W[(long)k * N + n];
      Wf[(long)k * N + n] = sa[k] * w;
      acc += sc[k] * w;
    }
    bf[n] = acc;
  }
}

// BN affine coefficients only (for epilogues that can't be folded)
__global__ void finalize_affine(const float* sum, const float* sq, float invRows,
                                const float* g, const float* be,
                                float* a, float* c, int N) {
  int tid = threadIdx.x;
  if (tid < N) {
    float m = sum[tid] * invRows;
    float v = fmaxf(sq[tid] * invRows - m * m, 0.f);
    float s = rsqrtf(v + BN_EPS) * g[tid];
    a[tid] = s;
    c[tid] = be[tid] - m * s;
  }
}

__device__ inline void atomicMaxF(float* addr, float v) {
  // Correct for all non-NaN floats (incl. -inf init): sign-aware int trick.
  if (v >= 0.f) atomicMax((int*)addr, __float_as_int(v));
  else          atomicMin((unsigned int*)addr, __float_as_uint(v));
}

// segment_max of (X*a2 + c2) into agg via float atomic max
__global__ __launch_bounds__(256)
void segmax_kernel(const float* __restrict__ X, const float* __restrict__ a,
                   const float* __restrict__ c, const int* __restrict__ edge0,
                   const int* __restrict__ edge1, int E, int N0,
                   float* __restrict__ agg, int H) {
  long gid = (long)blockIdx.x * blockDim.x + threadIdx.x;
  int Hq = H >> 2;                 // float4 per thread
  long e = gid / Hq;
  int j = (int)(gid - e * Hq);
  if (e >= 2L * E) return;
  int idx = (e < E) ? edge0[e] : (edge1[e - E] + N0);
  int cc = j * 4;
  v4f x  = *(const v4f*)&X[e * (long)H + cc];
  v4f av = *(const v4f*)&a[cc];
  v4f cv = *(const v4f*)&c[cc];
  float* dst = &agg[(long)idx * H + cc];
  atomicMaxF(dst + 0, x[0] * av[0] + cv[0]);
  atomicMaxF(dst + 1, x[1] * av[1] + cv[1]);
  atomicMaxF(dst + 2, x[2] * av[2] + cv[2]);
  atomicMaxF(dst + 3, x[3] * av[3] + cv[3]);
}

// out := out*a + c (final BN), in place
__global__ void apply_affine(float* out, const float* a, const float* c,
                             long n, int N) {
  long i = (long)blockIdx.x * blockDim.x + threadIdx.x;
  if (i < n) {
    int cc = (int)(i % N);
    out[i] = out[i] * a[cc] + c[cc];
  }
}

extern "C" void kernel_launch(void* const* d_in, const int* in_sizes, int n_in,
                              void* d_out, int out_size, void* d_ws, size_t ws_size,
                              hipStream_t stream) {
  // ---- inputs (setup_inputs dict order; params flattened W,b,g,be per layer) ----
  const float* lastC0 = (const float*)d_in[0];
  const float* lastF0 = (const float*)d_in[1];
  const float* curC0  = (const float*)d_in[2];
  const int*   edge0  = (const int*)d_in[3];
  const float* lastC1 = (const float*)d_in[4];
  const float* lastF1 = (const float*)d_in[5];
  const float* curC1  = (const float*)d_in[6];
  const int*   edge1  = (const int*)d_in[7];
  const float* W1 = (const float*)d_in[8],  *b1 = (const float*)d_in[9];
  const float* g1 = (const float*)d_in[10], *be1 = (const float*)d_in[11];
  const float* W2 = (const float*)d_in[12], *b2 = (const float*)d_in[13];
  const float* g2 = (const float*)d_in[14], *be2 = (const float*)d_in[15];
  const float* W3 = (const float*)d_in[16], *b3 = (const float*)d_in[17];
  const float* g3 = (const float*)d_in[18], *be3 = (const float*)d_in[19];
  const float* W4 = (const float*)d_in[20], *b4 = (const float*)d_in[21];
  const float* g4 = (const float*)d_in[22], *be4 = (const float*)d_in[23];

  const int N0    = in_sizes[2] / 3;      // 10000
  const int N1    = in_sizes[6] / 3;      // 10000
  const int NLAST = in_sizes[0] / 3;      // 20000
  const int E     = in_sizes[3] / 2;      // 160000
  const int F     = in_sizes[1] / NLAST;  // 125
  const int K1    = F + 3;                // 128
  const int H     = 256;
  const int OUTC  = in_sizes[21];         // 512
  const long rowsE = 2L * E;              // 320000
  const int  rowsO = N0 + N1;             // 20000

  // ---- workspace layout ----
  char* ws = (char*)d_ws;
  size_t off = 0;
  auto alloc = [&](size_t bytes) -> void* {
    void* p = ws + off;
    off = (off + bytes + 255) & ~(size_t)255;
    return p;
  };
  float* X1   = (float*)alloc(rowsE * H * 4);        // post layer-1 relu
  float* X2   = (float*)alloc(rowsE * H * 4);        // post layer-2 relu
  float* agg  = (float*)alloc((long)rowsO * H * 4);  // segment max (BN2 applied)
  float* X3   = (float*)alloc((long)rowsO * H * 4);  // post layer-3 relu
  float* W2f  = (float*)alloc((long)H * H * 4);
  float* b2f  = (float*)alloc(H * 4);
  float* W4f  = (float*)alloc((long)H * OUTC * 4);
  float* b4f  = (float*)alloc(OUTC * 4);
  const long statsN = 6L * H + 2L * OUTC;
  float* statsAll = (float*)alloc(statsN * 4);
  float* sum1 = statsAll;           float* sq1 = sum1 + H;
  float* sum2 = sq1 + H;            float* sq2 = sum2 + H;
  float* sum3 = sq2 + H;            float* sq3 = sum3 + H;
  float* sum4 = sq3 + H;            float* sq4 = sum4 + OUTC;
  float* a2 = (float*)alloc(H * 4);    float* c2 = (float*)alloc(H * 4);
  float* a4 = (float*)alloc(OUTC * 4); float* c4 = (float*)alloc(OUTC * 4);

  GatherArgs ga = {lastF0, lastC0, curC0, edge0,
                   lastF1, lastC1, curC1, edge1, E, F};
  GatherArgs gz = {};
  dim3 blk(256);

  // 1) init agg = -inf, stats = 0
  {
    long tot = (long)rowsO * H + statsN;
    init_kernel<<<(int)((tot + 255) / 256), blk, 0, stream>>>(
        (unsigned*)agg, (long)rowsO * H, statsAll, statsN);
  }
  // 2) gather + GEMM1 + relu + stats1
  gemm_relu_stats<MODE_GATHER><<<dim3((unsigned)((rowsE + TR - 1) / TR), H / TC),
                                 blk, 0, stream>>>(
      nullptr, W1, b1, X1, sum1, sq1, (int)rowsE, K1, H, ga);
  // 3) BN1 folded into W2
  finalize_fold<<<1, 256, 0, stream>>>(sum1, sq1, 1.0f / (float)rowsE, g1, be1,
                                       W2, b2, W2f, b2f, H, H);
  // 4) GEMM2 + relu + stats2
  gemm_relu_stats<MODE_PLAIN><<<dim3((unsigned)((rowsE + TR - 1) / TR), H / TC),
                                blk, 0, stream>>>(
      X1, W2f, b2f, X2, sum2, sq2, (int)rowsE, H, H, gz);
  // 5) BN2 affine coefficients
  finalize_affine<<<1, 256, 0, stream>>>(sum2, sq2, 1.0f / (float)rowsE,
                                         g2, be2, a2, c2, H);
  // 6) segment max of BN2(X2)
  {
    long tot = rowsE * (H / 4);
    segmax_kernel<<<(int)((tot + 255) / 256), blk, 0, stream>>>(
        X2, a2, c2, edge0, edge1, E, N0, agg, H);
  }
  // 7) GEMM3 (with -inf -> 0 fixup on load) + relu + stats3
  gemm_relu_stats<MODE_NEGINF><<<dim3((rowsO + TR - 1) / TR, H / TC),
                                 blk, 0, stream>>>(
      agg, W3, b3, X3, sum3, sq3, rowsO, H, H, gz);
  // 8) BN3 folded into W4
  finalize_fold<<<1, 256, 0, stream>>>(sum3, sq3, 1.0f / (float)rowsO, g3, be3,
                                       W4, b4, W4f, b4f, H, OUTC);
  // 9) GEMM4 + relu + stats4 -> raw output
  gemm_relu_stats<MODE_PLAIN><<<dim3((rowsO + TR - 1) / TR, OUTC / TC),
                                blk, 0, stream>>>(
      X3, W4f, b4f, (float*)d_out, sum4, sq4, rowsO, H, OUTC, gz);
  // 10) BN4 affine coefficients
  finalize_affine<<<1, 512, 0, stream>>>(sum4, sq4, 1.0f / (float)rowsO,
                                         g4, be4, a4, c4, OUTC);
  // 11) out := out*a4 + c4
  {
    long tot = (long)rowsO * OUTC;
    apply_affine<<<(int)((tot + 255) / 256), blk, 0, stream>>>(
        (float*)d_out, a4, c4, tot, OUTC);
  }
}